// GNNStack_43336220017044
// MI455X (gfx1250) — compile-verified
//
#include <hip/hip_runtime.h>
#include <hip/hip_bf16.h>
#include <math.h>

typedef __attribute__((ext_vector_type(2))) float v2f;
typedef __attribute__((ext_vector_type(8))) float v8f;

// ---------------------------------------------------------------------------
// WMMA fp32 GEMM:  C[M,N] = A[M,64] @ W[64,N] + bias[N]
// NT = number of 16-wide column tiles (compile-time so accumulators stay in
// statically-named VGPRs; a runtime bound previously forced v_movrel spills).
// One wave -> 16 rows x NT*16 cols, K stepped by 4 via V_WMMA_F32_16X16X4_F32.
// W staged in LDS zero-padded to 64x64. EXEC stays all-1s around the WMMAs.
// ---------------------------------------------------------------------------
template <int NT>
__global__ __launch_bounds__(256) void gemm_wmma_f32(
    const float* __restrict__ A, const float* __restrict__ W,
    const float* __restrict__ bias, float* __restrict__ C,
    int M, int N, int applyRelu)
{
    __shared__ float sW[64 * 64];
    __shared__ float sB[64];
    const int tid = threadIdx.x;

    for (int idx = tid; idx < 64 * 64; idx += 256) {
        int row = idx >> 6, col = idx & 63;
        sW[idx] = (col < N) ? W[row * N + col] : 0.0f;
    }
    if (tid < 64) sB[tid] = (tid < N) ? bias[tid] : 0.0f;
    __syncthreads();

    const int lane = tid & 31;
    const int wave = tid >> 5;
    const int hi   = lane >> 4;   // 0: lanes 0-15, 1: lanes 16-31
    const int lo   = lane & 15;
    const int rowBase = blockIdx.x * 128 + wave * 16;

    int arow = rowBase + lo;
    if (arow >= M) arow = M - 1;            // clamp; stores are guarded below
    const float* Arow = A + (size_t)arow * 64;

    v8f acc[NT];
    #pragma unroll
    for (int t = 0; t < NT; ++t)
        #pragma unroll
        for (int r = 0; r < 8; ++r) acc[t][r] = 0.0f;

    #pragma unroll
    for (int k = 0; k < 64; k += 4) {
        const int ka = k + 2 * hi;
        v2f a;                               // A 16x4 layout (ISA 7.12.2)
        a.x = Arow[ka];
        a.y = Arow[ka + 1];
        #pragma unroll
        for (int t = 0; t < NT; ++t) {
            const int col = t * 16 + lo;
            v2f b;                           // B 4x16, row striped across lanes
            b.x = sW[ka * 64 + col];
            b.y = sW[(ka + 1) * 64 + col];
            acc[t] = __builtin_amdgcn_wmma_f32_16x16x4_f32(
                false, a, false, b, (short)0, acc[t], false, false);
        }
    }

    #pragma unroll
    for (int t = 0; t < NT; ++t) {
        const int col = t * 16 + lo;
        if (col >= N) continue;
        const float bv = sB[col];
        #pragma unroll
        for (int r = 0; r < 8; ++r) {
            const int orow = rowBase + 8 * hi + r;   // C/D layout: VGPR r -> M=r(+8)
            if (orow < M) {
                float v = acc[t][r] + bv;
                if (applyRelu) v = v > 0.0f ? v : 0.0f;
                C[(size_t)orow * N + col] = v;
            }
        }
    }
}

static void launch_gemm(const float* A, const float* W, const float* bias,
                        float* C, int M, int N, int applyRelu, hipStream_t stream)
{
    dim3 blk(256);
    dim3 grd((M + 127) / 128);
    const int nt = (N + 15) >> 4;
    switch (nt) {
        case 1: gemm_wmma_f32<1><<<grd, blk, 0, stream>>>(A, W, bias, C, M, N, applyRelu); break;
        case 2: gemm_wmma_f32<2><<<grd, blk, 0, stream>>>(A, W, bias, C, M, N, applyRelu); break;
        case 3: gemm_wmma_f32<3><<<grd, blk, 0, stream>>>(A, W, bias, C, M, N, applyRelu); break;
        default: gemm_wmma_f32<4><<<grd, blk, 0, stream>>>(A, W, bias, C, M, N, applyRelu); break;
    }
}

// s_dst[i] = h[i,:] . att[0:64] ; s_src[i] = h[i,:] . att[64:128]
__global__ __launch_bounds__(256) void node_scores(
    const float* __restrict__ h, const float* __restrict__ att,
    float* __restrict__ sdst, float* __restrict__ ssrc, int M)
{
    const int node = blockIdx.x * 8 + (threadIdx.x >> 5);
    const int lane = threadIdx.x & 31;
    if (node >= M) return;
    const float* hr = h + (size_t)node * 64;
    const float h0 = hr[lane], h1 = hr[lane + 32];
    float pd = h0 * att[lane]      + h1 * att[lane + 32];
    float ps = h0 * att[64 + lane] + h1 * att[96 + lane];
    for (int off = 16; off >= 1; off >>= 1) {
        pd += __shfl_xor(pd, off, 32);
        ps += __shfl_xor(ps, off, 32);
    }
    if (lane == 0) { sdst[node] = pd; ssrc[node] = ps; }
}

__device__ __forceinline__ unsigned enc_f32(float f) {
    unsigned b = __float_as_uint(f);
    return (b & 0x80000000u) ? ~b : (b | 0x80000000u);   // order-preserving
}
__device__ __forceinline__ float dec_f32(unsigned u) {
    unsigned b = (u & 0x80000000u) ? (u & 0x7fffffffu) : ~u;
    return __uint_as_float(b);
}

// e = leaky_relu(s_dst[dst] + s_src[src], 0.2);  m[dst] = segment_max(e)
__global__ void edge_leaky_max(
    const long long* __restrict__ srcI, const long long* __restrict__ dstI,
    const float* __restrict__ sdst, const float* __restrict__ ssrc,
    float* __restrict__ ebuf, unsigned* __restrict__ mEnc, int NE)
{
    const int i = blockIdx.x * blockDim.x + threadIdx.x;
    if (i >= NE) return;
    const int s = (int)srcI[i], d = (int)dstI[i];
    const float v = sdst[d] + ssrc[s];
    const float e = v > 0.0f ? v : 0.2f * v;
    ebuf[i] = e;
    atomicMax(&mEnc[d], enc_f32(e));
}

// ex = exp(e - m[dst]); denom[dst] += ex   (ex overwrites ebuf)
__global__ void edge_exp_sum(
    const long long* __restrict__ dstI, const unsigned* __restrict__ mEnc,
    float* __restrict__ ebuf, float* __restrict__ denom, int NE)
{
    const int i = blockIdx.x * blockDim.x + threadIdx.x;
    if (i >= NE) return;
    const int d = (int)dstI[i];
    const float ex = __expf(ebuf[i] - dec_f32(mEnc[d]));
    ebuf[i] = ex;
    atomicAdd(&denom[d], ex);
}

// agg[dst,:] += h[src,:] * (ex / denom[dst]) ; one wave per edge, 2 cols/lane
__global__ __launch_bounds__(256) void edge_aggregate(
    const long long* __restrict__ srcI, const long long* __restrict__ dstI,
    const float* __restrict__ h, const float* __restrict__ ebuf,
    const float* __restrict__ denom, float* __restrict__ agg, int NE)
{
    const int e = blockIdx.x * 8 + (threadIdx.x >> 5);
    const int lane = threadIdx.x & 31;
    if (e >= NE) return;
    const int s = (int)srcI[e], d = (int)dstI[e];
    const float alpha = ebuf[e] / denom[d];
    const float* hs = h + (size_t)s * 64;
    float* ad = agg + (size_t)d * 64;
    atomicAdd(&ad[lane],      hs[lane]      * alpha);
    atomicAdd(&ad[lane + 32], hs[lane + 32] * alpha);
}

// out = (relu?) (agg + bias)
__global__ void bias_act(const float* __restrict__ agg,
                         const float* __restrict__ bias,
                         float* __restrict__ out, int total, int applyRelu)
{
    const int i = blockIdx.x * blockDim.x + threadIdx.x;
    if (i >= total) return;
    float v = agg[i] + bias[i & 63];
    if (applyRelu) v = fmaxf(v, 0.0f);
    out[i] = v;
}

// in-place row-wise log_softmax for N <= 64 (one wave per row)
__global__ __launch_bounds__(256) void log_softmax_rows(
    float* __restrict__ out, int M, int N)
{
    const int node = blockIdx.x * 8 + (threadIdx.x >> 5);
    const int lane = threadIdx.x & 31;
    if (node >= M) return;
    float* row = out + (size_t)node * N;
    const int c2 = lane + 32;
    const float v1 = (lane < N) ? row[lane] : -INFINITY;
    const float v2 = (c2  < N) ? row[c2]   : -INFINITY;
    float mx = fmaxf(v1, v2);
    for (int off = 16; off >= 1; off >>= 1) mx = fmaxf(mx, __shfl_xor(mx, off, 32));
    float s = ((lane < N) ? __expf(v1 - mx) : 0.0f) +
              ((c2  < N) ? __expf(v2 - mx) : 0.0f);
    for (int off = 16; off >= 1; off >>= 1) s += __shfl_xor(s, off, 32);
    const float ls = __logf(s);
    if (lane < N) row[lane] = v1 - mx - ls;
    if (c2  < N) row[c2]   = v2 - mx - ls;
}

// ---------------------------------------------------------------------------
static void run_gat_layer(const float* in, const float* W, const float* b,
                          const float* att, const float* bias,
                          float* hbuf, float* outbuf,
                          float* ebuf, float* sdst, float* ssrc,
                          unsigned* mEnc, float* denom, float* agg,
                          const long long* srcI, const long long* dstI,
                          int M, int NE, hipStream_t stream)
{
    dim3 blk(256);
    launch_gemm(in, W, b, hbuf, M, 64, 0, stream);
    node_scores<<<(M + 7) / 8, blk, 0, stream>>>(hbuf, att, sdst, ssrc, M);
    hipMemsetAsync(mEnc, 0, (size_t)M * sizeof(unsigned), stream);
    hipMemsetAsync(denom, 0, (size_t)M * sizeof(float), stream);
    hipMemsetAsync(agg, 0, (size_t)M * 64 * sizeof(float), stream);
    edge_leaky_max<<<(NE + 255) / 256, blk, 0, stream>>>(srcI, dstI, sdst, ssrc, ebuf, mEnc, NE);
    edge_exp_sum<<<(NE + 255) / 256, blk, 0, stream>>>(dstI, mEnc, ebuf, denom, NE);
    edge_aggregate<<<(NE + 7) / 8, blk, 0, stream>>>(srcI, dstI, hbuf, ebuf, denom, agg, NE);
    bias_act<<<(M * 64 + 255) / 256, blk, 0, stream>>>(agg, bias, outbuf, M * 64, 1);
}

extern "C" void kernel_launch(void* const* d_in, const int* in_sizes, int n_in,
                              void* d_out, int out_size, void* d_ws, size_t ws_size,
                              hipStream_t stream)
{
    const float*     x     = (const float*)d_in[0];
    const long long* ei    = (const long long*)d_in[1];   // int64, shape (2, NE)
    const float*     W0    = (const float*)d_in[2];
    const float*     b0    = (const float*)d_in[3];
    const float*     att0  = (const float*)d_in[4];
    const float*     bias0 = (const float*)d_in[5];
    const float*     W1    = (const float*)d_in[6];
    const float*     b1    = (const float*)d_in[7];
    const float*     att1  = (const float*)d_in[8];
    const float*     bias1 = (const float*)d_in[9];
    const float*     Wp1   = (const float*)d_in[10];
    const float*     bp1   = (const float*)d_in[11];
    const float*     Wp2   = (const float*)d_in[12];
    const float*     bp2   = (const float*)d_in[13];

    const int M    = in_sizes[0] / 64;      // 100000 nodes
    const int NE   = in_sizes[1] / 2;       // 1000000 edges
    const int DOUT = in_sizes[12] / 64;     // 40

    const long long* srcI = ei;
    const long long* dstI = ei + NE;

    // workspace layout (256B aligned slices)
    char* ws = (char*)d_ws;
    size_t off = 0;
    auto take = [&](size_t bytes) -> char* {
        char* p = ws + off;
        off += (bytes + 255) & ~(size_t)255;
        return p;
    };
    float*    h0    = (float*)take((size_t)M * 64 * sizeof(float));
    float*    h1    = (float*)take((size_t)M * 64 * sizeof(float));
    float*    agg   = (float*)take((size_t)M * 64 * sizeof(float));
    float*    ebuf  = (float*)take((size_t)NE * sizeof(float));
    float*    sdst  = (float*)take((size_t)M * sizeof(float));
    float*    ssrc  = (float*)take((size_t)M * sizeof(float));
    unsigned* mEnc  = (unsigned*)take((size_t)M * sizeof(unsigned));
    float*    denom = (float*)take((size_t)M * sizeof(float));

    // GAT layer 0: x -> h1 (relu'd)
    run_gat_layer(x,  W0, b0, att0, bias0, h0, h1, ebuf, sdst, ssrc, mEnc, denom, agg,
                  srcI, dstI, M, NE, stream);
    // GAT layer 1: h1 -> h1 (relu'd), using h0 as the layer's h scratch
    run_gat_layer(h1, W1, b1, att1, bias1, h0, h1, ebuf, sdst, ssrc, mEnc, denom, agg,
                  srcI, dstI, M, NE, stream);

    // MLP head: p1 = h1@Wp1+bp1 ; logits = p1@Wp2+bp2 ; log_softmax
    launch_gemm(h1, Wp1, bp1, h0, M, 64, 0, stream);
    launch_gemm(h0, Wp2, bp2, (float*)d_out, M, DOUT, 0, stream);
    log_softmax_rows<<<(M + 7) / 8, dim3(256), 0, stream>>>((float*)d_out, M, DOUT);
}